// DiffeomorphicTransform_38104949850356
// MI455X (gfx1250) — compile-verified
//
#include <hip/hip_runtime.h>
#include <hip/hip_bf16.h>
#include <stdint.h>

// ---------------------------------------------------------------------------
// Diffeomorphic transform (scaling-and-squaring), [B=32, 2, H=768, W=768] f32.
// 7 ping-pong launches of:  dst = s*src + s*bilinear(src, id + s*src)
// (s = 1/128 on the first launch folds flow0 = v/2^7 into iteration 1).
// Each workgroup stages an 82x82-per-channel source window into LDS via the
// CDNA5 Tensor Data Mover (tensor_load_to_lds, TENSORcnt), then samples
// bilinearly from LDS with a bounds-checked global fallback for the rare
// out-of-halo coordinate (zeros outside the image).
// TILE=64 => halo read amplification 82^2/64^2 = 1.64x (vs 2.44x at TILE=32);
// LDS 53.8KB/block -> 5 blocks per 320KB WGP.
// ---------------------------------------------------------------------------

#define IMG_W   768
#define IMG_H   768
#define TILE    64
#define HALO    8
#define SRCW    (TILE + 2 * HALO + 2)   // 82 (TDM fills rows contiguously at this pitch)
#define SRCH    (TILE + 2 * HALO + 2)   // 82

typedef __attribute__((ext_vector_type(4))) unsigned int u32x4;
typedef __attribute__((ext_vector_type(8))) int          i32x8;
typedef __attribute__((ext_vector_type(4))) int          i32x4;

#if defined(__gfx1250__) && __has_builtin(__builtin_amdgcn_tensor_load_to_lds) && \
    __has_builtin(__builtin_amdgcn_s_wait_tensorcnt)
#define USE_TDM 1
#else
#define USE_TDM 0
#endif

__device__ __forceinline__ void fetch_corner(const float* __restrict__ sm0,
                                             const float* __restrict__ sm1,
                                             const float* __restrict__ gbase,
                                             int x0, int y0, int iy, int ix,
                                             float& vy, float& vx) {
    const int sy = iy - y0;
    const int sx = ix - x0;
    if ((unsigned)sy < (unsigned)SRCH && (unsigned)sx < (unsigned)SRCW) {
        const int o = sy * SRCW + sx;           // LDS hit (common case)
        vy = sm0[o];
        vx = sm1[o];
    } else if ((unsigned)iy < (unsigned)IMG_H && (unsigned)ix < (unsigned)IMG_W) {
        const size_t o = (size_t)iy * IMG_W + ix;   // rare: outside halo window
        vy = gbase[o];
        vx = gbase[(size_t)IMG_H * IMG_W + o];
    } else {                                        // outside image: zero pad
        vy = 0.0f;
        vx = 0.0f;
    }
}

__global__ void __launch_bounds__(256)
diffeo_warp_step(const float* __restrict__ src, float* __restrict__ dst,
                 float scale) {
    __shared__ float sm[2][SRCH * SRCW];    // 2 * 82*82 * 4B = 53,792 B

    const int tx = blockIdx.x * TILE;
    const int ty = blockIdx.y * TILE;
    const int b  = blockIdx.z;

    // Source window origin, clamped fully inside the image so the TDM tile
    // never needs OOB handling; tile pixels always land inside the window.
    int x0 = tx - HALO; x0 = x0 < 0 ? 0 : x0; x0 = x0 > (IMG_W - SRCW) ? (IMG_W - SRCW) : x0;
    int y0 = ty - HALO; y0 = y0 < 0 ? 0 : y0; y0 = y0 > (IMG_H - SRCH) ? (IMG_H - SRCH) : y0;

    const float* gbase = src + (size_t)b * 2 * IMG_H * IMG_W;

#if USE_TDM
    if (threadIdx.y == 0) {                 // wave 0 only: one TDM issue per channel
        #pragma unroll
        for (int c = 0; c < 2; ++c) {
            const uint64_t ga = (uint64_t)(uintptr_t)
                (gbase + (size_t)c * IMG_H * IMG_W + (size_t)y0 * IMG_W + x0);
            const uint32_t la = (uint32_t)(uintptr_t)(&sm[c][0]);   // LDS byte offset

            // D# group 0: count=1 | lds_addr | global_addr[56:0] | type=2
            u32x4 g0 = { 1u,
                         la,
                         (uint32_t)ga,
                         (uint32_t)((ga >> 32) & 0x01FFFFFFu) | 0x80000000u };

            // D# group 1: data_size=4B; tensor 768x768; tile 82x82; stride 768
            i32x8 g1 = { (int)0x00020000,                                  // data_size=2 (4B)
                         (int)((IMG_W & 0xFFFF) << 16),                    // tensor_dim0[15:0]
                         (int)(((IMG_W >> 16) & 0xFFFF) |
                               ((IMG_H & 0xFFFF) << 16)),                  // dim0 hi | dim1 lo
                         (int)(((IMG_H >> 16) & 0xFFFF) | (SRCW << 16)),   // dim1 hi | tile_dim0
                         (int)SRCH,                                        // tile_dim1 (tile_dim2=0)
                         (int)IMG_W,                                       // tensor_dim0_stride lo
                         0, 0 };

            i32x4 gz4 = { 0, 0, 0, 0 };         // groups 2/3 unused (2-D tensor)
            i32x8 gz8 = { 0, 0, 0, 0, 0, 0, 0, 0 };
            // 6-arg form (amdgpu-toolchain clang-23 / therock-10.0 headers)
            __builtin_amdgcn_tensor_load_to_lds(g0, g1, gz4, gz4, gz8, 0);
        }
        __builtin_amdgcn_s_wait_tensorcnt(0);
    }
#else
    // Portable fallback: cooperative global->LDS copy.
    for (int c = 0; c < 2; ++c) {
        const float* gp = gbase + (size_t)c * IMG_H * IMG_W;
        for (int i = threadIdx.y * 32 + threadIdx.x; i < SRCH * SRCW; i += 256) {
            const int ry = i / SRCW, rx = i - ry * SRCW;
            sm[c][i] = gp[(size_t)(y0 + ry) * IMG_W + (x0 + rx)];
        }
    }
#endif
    __syncthreads();

    const float* sm0 = &sm[0][0];
    const float* sm1 = &sm[1][0];
    const size_t plane = (size_t)IMG_H * IMG_W;
    float* dsty = dst + (size_t)b * 2 * plane;
    float* dstx = dsty + plane;

    // Block is 32x8 (one wave per row-slab); each lane covers 2 columns and
    // 8 rows of the 64x64 tile.
    for (int cseg = 0; cseg < 2; ++cseg) {
        const int gx = tx + cseg * 32 + threadIdx.x;
        for (int r = 0; r < TILE / 8; ++r) {
            const int gy = ty + threadIdx.y + r * 8;
            const int so = (gy - y0) * SRCW + (gx - x0);

            const float fy = scale * sm0[so];
            const float fx = scale * sm1[so];
            const float cy = (float)gy + fy;
            const float cx = (float)gx + fx;

            const float yf = floorf(cy);
            const float xf = floorf(cx);
            const float wy = cy - yf;
            const float wx = cx - xf;
            const int iy = (int)yf;
            const int ix = (int)xf;

            float v00y, v00x, v01y, v01x, v10y, v10x, v11y, v11x;
            fetch_corner(sm0, sm1, gbase, x0, y0, iy,     ix,     v00y, v00x);
            fetch_corner(sm0, sm1, gbase, x0, y0, iy,     ix + 1, v01y, v01x);
            fetch_corner(sm0, sm1, gbase, x0, y0, iy + 1, ix,     v10y, v10x);
            fetch_corner(sm0, sm1, gbase, x0, y0, iy + 1, ix + 1, v11y, v11x);

            const float w00 = (1.0f - wy) * (1.0f - wx);
            const float w01 = (1.0f - wy) * wx;
            const float w10 = wy * (1.0f - wx);
            const float w11 = wy * wx;

            const float sy = w00 * v00y + w01 * v01y + w10 * v10y + w11 * v11y;
            const float sx = w00 * v00x + w01 * v01x + w10 * v10x + w11 * v11x;

            const size_t oo = (size_t)gy * IMG_W + gx;
            dsty[oo] = fy + scale * sy;     // flow + flow∘(id+flow)
            dstx[oo] = fx + scale * sx;
        }
    }
}

extern "C" void kernel_launch(void* const* d_in, const int* in_sizes, int n_in,
                              void* d_out, int out_size, void* d_ws, size_t ws_size,
                              hipStream_t stream) {
    (void)in_sizes; (void)n_in; (void)out_size; (void)ws_size;
    const float* vel = (const float*)d_in[0];
    float*       out = (float*)d_out;
    float*       ws  = (float*)d_ws;

    const dim3 grid(IMG_W / TILE, IMG_H / TILE, 32);   // 12 x 12 x 32
    const dim3 block(32, 8);
    const float s0 = 1.0f / 128.0f;     // 2^-TIME_STEP folded into iteration 1

    // 7 squaring steps, ping-pong so the final result lands in d_out.
    diffeo_warp_step<<<grid, block, 0, stream>>>(vel, out, s0);    // 1: vel -> out
    diffeo_warp_step<<<grid, block, 0, stream>>>(out, ws,  1.0f);  // 2
    diffeo_warp_step<<<grid, block, 0, stream>>>(ws,  out, 1.0f);  // 3
    diffeo_warp_step<<<grid, block, 0, stream>>>(out, ws,  1.0f);  // 4
    diffeo_warp_step<<<grid, block, 0, stream>>>(ws,  out, 1.0f);  // 5
    diffeo_warp_step<<<grid, block, 0, stream>>>(out, ws,  1.0f);  // 6
    diffeo_warp_step<<<grid, block, 0, stream>>>(ws,  out, 1.0f);  // 7: -> out
}